// GraphSAGE_42245298324238
// MI455X (gfx1250) — compile-verified
//
#include <hip/hip_runtime.h>
#include <hip/hip_bf16.h>

#define N_NODES 50000   // multiple of 16 -> no tail handling in WMMA kernel
#define N_EDGES 800000
#define FEATS   128

typedef __attribute__((ext_vector_type(2))) float v2f;
typedef __attribute__((ext_vector_type(8))) float v8f;

// ---------------------------------------------------------------------------
// Zero a float range.
__global__ void zero_kernel(float* __restrict__ p, long long n) {
    long long i = (long long)blockIdx.x * blockDim.x + threadIdx.x;
    long long stride = (long long)gridDim.x * blockDim.x;
    for (; i < n; i += stride) p[i] = 0.0f;
}

// ---------------------------------------------------------------------------
// deg[dst[e]] += 1.0f  (f32 atomic; resolved in L2 RMW units)
__global__ void degree_kernel(const int* __restrict__ dst,
                              float* __restrict__ deg, int nEdges) {
    int e = blockIdx.x * blockDim.x + threadIdx.x;
    if (e < nEdges) atomicAdd(&deg[dst[e]], 1.0f);
}

// ---------------------------------------------------------------------------
// agg[dst[e], :] += x[src[e], :]
// One wave per edge: 32 lanes x float4 = 512B coalesced gather of the source
// row, then 128 contiguous global_atomic_add_f32 into the destination row.
__global__ void scatter_add_kernel(const float* __restrict__ X,
                                   const int* __restrict__ src,
                                   const int* __restrict__ dst,
                                   float* __restrict__ agg, int nEdges) {
    long long t = (long long)blockIdx.x * blockDim.x + threadIdx.x;
    int e = (int)(t >> 5);
    if (e >= nEdges) return;
    int c = ((int)t & 31) * 4;
    int s = src[e];
    int d = dst[e];
    const float4 v = *(const float4*)(X + (size_t)s * FEATS + c);
    float* p = agg + (size_t)d * FEATS + c;
    atomicAdd(p + 0, v.x);
    atomicAdd(p + 1, v.y);
    atomicAdd(p + 2, v.z);
    atomicAdd(p + 3, v.w);
}

// ---------------------------------------------------------------------------
// OUT[m, n] = act( X[m,:] @ Wself + (AGG[m,:]/max(deg,1)) @ Wneigh + bias[n] )
//
// Block = 256 threads = 8 waves; block handles a 16-row node tile, wave w
// computes the 16x16 output tile at columns [16w, 16w+16) with
// V_WMMA_F32_16X16X4_F32 (f32-exact, 64 WMMAs per wave).
//
// The 16x128 X and AGG tiles (16 KB) are staged into LDS once per block with
// coalesced float4 loads (AGG pre-scaled by 1/deg), so the 8 waves read their
// A fragments from LDS (ds_load_b64) instead of re-reading global 8x over.
//
// Fragment layouts (ISA 7.12.2, 32-bit, wave32):
//   A 16x4: lanes 0-15 M=lane   K={k,k+1};  lanes 16-31 M=lane-16 K={k+2,k+3}
//   B 4x16: lanes 0-15 N=lane rows{k,k+1};  lanes 16-31 N=lane-16 rows{k+2,k+3}
//   C 16x16: vgpr r -> lanes 0-15 M=r, N=lane; lanes 16-31 M=r+8, N=lane-16
template <bool RELU>
__global__ void sage_wmma_kernel(const float* __restrict__ X,
                                 const float* __restrict__ AGG,
                                 const float* __restrict__ DEG,
                                 const float* __restrict__ Wself,
                                 const float* __restrict__ Wneigh,
                                 const float* __restrict__ bias,
                                 float* __restrict__ OUT) {
    __shared__ float sInvd[16];
    __shared__ float sX[16 * FEATS];
    __shared__ float sA[16 * FEATS];

    const int tid  = threadIdx.x;
    const int lane = tid & 31;
    const int wave = tid >> 5;               // 0..7 -> N tile
    const int half = lane >> 4;              // 0: lanes 0-15, 1: lanes 16-31
    const int l15  = lane & 15;
    const int node0 = blockIdx.x * 16;
    const int nOff  = wave * 16;

    if (tid < 16) sInvd[tid] = 1.0f / fmaxf(DEG[node0 + tid], 1.0f);
    __syncthreads();

    // Stage 16x128 tiles: 512 float4 slots, 2 per thread, fully coalesced.
    #pragma unroll
    for (int i = tid; i < 16 * 32; i += 256) {
        const int r  = i >> 5;
        const int c4 = (i & 31) * 4;
        const size_t off = (size_t)(node0 + r) * FEATS + c4;
        const float4 vx = *(const float4*)(X + off);
        const float4 va = *(const float4*)(AGG + off);
        const float inv = sInvd[r];
        *(float4*)(sX + r * FEATS + c4) = vx;
        float4 vs;
        vs.x = va.x * inv; vs.y = va.y * inv;
        vs.z = va.z * inv; vs.w = va.w * inv;
        *(float4*)(sA + r * FEATS + c4) = vs;
    }
    __syncthreads();

    const float* xrow = sX + l15 * FEATS;    // this lane's A row in LDS
    const float* arow = sA + l15 * FEATS;

    v8f c = {0.f, 0.f, 0.f, 0.f, 0.f, 0.f, 0.f, 0.f};

    #pragma unroll 4
    for (int k = 0; k < FEATS; k += 4) {
        const int ka = k + half * 2;                      // lane's K pair (8B aligned)
        const v2f aS = *(const v2f*)(xrow + ka);          // ds_load_b64
        const v2f aN = *(const v2f*)(arow + ka);

        const float* wS = Wself  + (size_t)ka * FEATS + nOff + l15;
        const float* wN = Wneigh + (size_t)ka * FEATS + nOff + l15;
        v2f bS; bS.x = wS[0]; bS.y = wS[FEATS];
        v2f bN; bN.x = wN[0]; bN.y = wN[FEATS];

        c = __builtin_amdgcn_wmma_f32_16x16x4_f32(false, aS, false, bS,
                                                  (short)0, c, false, false);
        c = __builtin_amdgcn_wmma_f32_16x16x4_f32(false, aN, false, bN,
                                                  (short)0, c, false, false);
    }

    // Branch-free epilogue: N_NODES % 16 == 0, RELU resolved at compile time.
    const float bv = bias[nOff + l15];
    float* orow = OUT + (size_t)(node0 + half * 8) * FEATS + nOff + l15;
    #pragma unroll
    for (int r = 0; r < 8; ++r) {
        float v = c[r] + bv;
        if (RELU) v = fmaxf(v, 0.0f);
        orow[(size_t)r * FEATS] = v;
    }
}

// ---------------------------------------------------------------------------
extern "C" void kernel_launch(void* const* d_in, const int* in_sizes, int n_in,
                              void* d_out, int out_size, void* d_ws, size_t ws_size,
                              hipStream_t stream) {
    const float* x        = (const float*)d_in[0];
    const int*   src      = (const int*)  d_in[1];
    const int*   dst      = (const int*)  d_in[2];
    const float* W_self1  = (const float*)d_in[3];
    const float* W_neigh1 = (const float*)d_in[4];
    const float* b1       = (const float*)d_in[5];
    const float* W_self2  = (const float*)d_in[6];
    const float* W_neigh2 = (const float*)d_in[7];
    const float* b2       = (const float*)d_in[8];
    float* out = (float*)d_out;

    // workspace layout: deg | agg | h1  (~51.5 MB)
    float* deg = (float*)d_ws;
    float* agg = deg + N_NODES;
    float* h1  = agg + (size_t)N_NODES * FEATS;

    const int ZB = 256;
    const long long degAgg = (long long)N_NODES * (FEATS + 1);
    const long long scatterThreads = (long long)N_EDGES * 32;
    const int scatterBlocks = (int)((scatterThreads + ZB - 1) / ZB);
    const int tiles = N_NODES / 16;                      // 3125, exact

    // ---- layer 1 ----
    zero_kernel<<<2048, ZB, 0, stream>>>(deg, degAgg);   // deg + agg contiguous
    degree_kernel<<<(N_EDGES + ZB - 1) / ZB, ZB, 0, stream>>>(dst, deg, N_EDGES);
    scatter_add_kernel<<<scatterBlocks, ZB, 0, stream>>>(x, src, dst, agg, N_EDGES);
    sage_wmma_kernel<true><<<tiles, 256, 0, stream>>>(x, agg, deg, W_self1,
                                                      W_neigh1, b1, h1);

    // ---- layer 2 ----
    zero_kernel<<<2048, ZB, 0, stream>>>(agg, (long long)N_NODES * FEATS);
    scatter_add_kernel<<<scatterBlocks, ZB, 0, stream>>>(h1, src, dst, agg, N_EDGES);
    sage_wmma_kernel<false><<<tiles, 256, 0, stream>>>(h1, agg, deg, W_self2,
                                                       W_neigh2, b2, out);
}